// GAT_Protein_60395830116947
// MI455X (gfx1250) — compile-verified
//
#include <hip/hip_runtime.h>
#include <math.h>

typedef __attribute__((ext_vector_type(2))) float v2f;
typedef __attribute__((ext_vector_type(8))) float v8f;

#define HDIM 128
#define ALPHA 0.2f

__device__ __forceinline__ float leaky(float v) { return v >= 0.f ? v : ALPHA * v; }

// Order-preserving float <-> uint map so unsigned atomicMax == float max.
__device__ __forceinline__ unsigned enc_f32(float f) {
    unsigned u = __float_as_uint(f);
    return (u & 0x80000000u) ? ~u : (u | 0x80000000u);
}
__device__ __forceinline__ float dec_f32(unsigned u) {
    unsigned b = (u & 0x80000000u) ? (u & 0x7FFFFFFFu) : ~u;
    return __uint_as_float(b);
}
#define ENC_NEG_INF 0x007FFFFFu   // enc_f32(-inf)

// ---------------------------------------------------------------------------
// fp32 WMMA GEMM:  C[M x 128] = act(A[M x K] @ B[K x 128] + bias)
// 256 threads = 8 waves; each wave owns a 32-row strip (two 16-row WMMA tiles)
// and all 8 column tiles, so every B fragment feeds 2 WMMAs (16 wmma per
// 18 VMEM ops per k-step).  V_WMMA_F32_16X16X4_F32, K stepped by 4.
// act: 0 = identity, 1 = relu
// ---------------------------------------------------------------------------
__global__ __launch_bounds__(256) void gemm_wmma_f32(
    const float* __restrict__ A, const float* __restrict__ B,
    const float* __restrict__ bias, float* __restrict__ C,
    int M, int K, int act)
{
    const int lane = threadIdx.x & 31;
    const int wib  = threadIdx.x >> 5;
    const int wave = blockIdx.x * 8 + wib;
    const int rt0  = wave * 2;                       // first 16-row tile
    const int rt1  = wave * 2 + 1;                   // second 16-row tile
    if (rt0 * 16 >= M) return;                       // uniform per wave

    const int mlocal = lane & 15;
    const int koff   = (lane < 16) ? 0 : 2;
    int m0 = rt0 * 16 + mlocal;
    int m1 = rt1 * 16 + mlocal;
    int mA0 = m0 < M ? m0 : (M - 1);                 // clamp tail loads
    int mA1 = m1 < M ? m1 : (M - 1);
    const float* Arow0 = A + (size_t)mA0 * K + koff;
    const float* Arow1 = A + (size_t)mA1 * K + koff;

    v8f acc0[8], acc1[8];
#pragma unroll
    for (int j = 0; j < 8; ++j) { acc0[j] = (v8f)(0.f); acc1[j] = (v8f)(0.f); }

    for (int k = 0; k < K; k += 4) {
        v2f a0 = *(const v2f*)(Arow0 + k);           // A[m][k+koff .. +1]
        v2f a1 = *(const v2f*)(Arow1 + k);
        const float* B0 = B + (size_t)(k + koff) * HDIM + mlocal;
#pragma unroll
        for (int j = 0; j < 8; ++j) {
            v2f b;
            b.x = B0[j * 16];                        // B[k+koff  ][j*16+n]
            b.y = B0[HDIM + j * 16];                 // B[k+koff+1][j*16+n]
            acc0[j] = __builtin_amdgcn_wmma_f32_16x16x4_f32(
                false, a0, false, b, (short)0, acc0[j], false, false);
            acc1[j] = __builtin_amdgcn_wmma_f32_16x16x4_f32(
                false, a1, false, b, (short)0, acc1[j], false, false);
        }
    }

    const int rowOff = (lane < 16) ? 0 : 8;
#pragma unroll
    for (int j = 0; j < 8; ++j) {
        int n = j * 16 + mlocal;
        float bv = bias[n];
#pragma unroll
        for (int r = 0; r < 8; ++r) {
            int row0 = rt0 * 16 + r + rowOff;
            if (row0 < M) {
                float v = acc0[j][r] + bv;
                if (act == 1) v = v > 0.f ? v : 0.f;
                C[(size_t)row0 * HDIM + n] = v;
            }
            int row1 = rt1 * 16 + r + rowOff;
            if (row1 < M) {
                float v = acc1[j][r] + bv;
                if (act == 1) v = v > 0.f ? v : 0.f;
                C[(size_t)row1 * HDIM + n] = v;
            }
        }
    }
}

// ---------------------------------------------------------------------------
// Edge phase. e < E -> (src=ei[e], dst=ei[E+e]); e >= E -> self loop (e-E).
// xl/xr are 2.56 MB each -> all gathers hit L2 (192 MB).
// ---------------------------------------------------------------------------
__global__ void edge_score_k(const float* __restrict__ xl, const float* __restrict__ xr,
                             const float* __restrict__ att, const long long* __restrict__ ei,
                             int E, int N, float* __restrict__ score,
                             unsigned* __restrict__ maxb)
{
    int wid  = (int)((blockIdx.x * blockDim.x + threadIdx.x) >> 5);
    int lane = threadIdx.x & 31;
    int ET = E + N;
    if (wid >= ET) return;
    long long s, d;
    if (wid < E) { s = ei[wid]; d = ei[E + wid]; } else { s = d = (long long)(wid - E); }

    float4 a = ((const float4*)(xl + (size_t)s * HDIM))[lane];
    float4 b = ((const float4*)(xr + (size_t)d * HDIM))[lane];
    float4 w = ((const float4*)att)[lane];
    float p = leaky(a.x + b.x) * w.x + leaky(a.y + b.y) * w.y +
              leaky(a.z + b.z) * w.z + leaky(a.w + b.w) * w.w;
#pragma unroll
    for (int off = 16; off > 0; off >>= 1) p += __shfl_down(p, off, 32);
    if (lane == 0) {
        score[wid] = p;
        atomicMax(&maxb[d], enc_f32(p));
    }
}

__global__ void edge_exp_k(const float* __restrict__ score, const long long* __restrict__ ei,
                           int E, int N, const unsigned* __restrict__ maxb,
                           float* __restrict__ ex, float* __restrict__ sum)
{
    int e = blockIdx.x * blockDim.x + threadIdx.x;
    int ET = E + N;
    if (e >= ET) return;
    long long d = (e < E) ? ei[E + e] : (long long)(e - E);
    float v = expf(score[e] - dec_f32(maxb[d]));
    ex[e] = v;
    atomicAdd(&sum[d], v);
}

__global__ void edge_aggr_k(const float* __restrict__ xl, const float* __restrict__ ex,
                            const float* __restrict__ sum, const long long* __restrict__ ei,
                            int E, int N, float* __restrict__ gat)
{
    int wid  = (int)((blockIdx.x * blockDim.x + threadIdx.x) >> 5);
    int lane = threadIdx.x & 31;
    int ET = E + N;
    if (wid >= ET) return;
    long long s, d;
    if (wid < E) { s = ei[wid]; d = ei[E + wid]; } else { s = d = (long long)(wid - E); }
    float a = ex[wid] / sum[d];
    float4 v = ((const float4*)(xl + (size_t)s * HDIM))[lane];
    float* dst = gat + (size_t)d * HDIM + lane * 4;
    atomicAdd(dst + 0, a * v.x);
    atomicAdd(dst + 1, a * v.y);
    atomicAdd(dst + 2, a * v.z);
    atomicAdd(dst + 3, a * v.w);
}

__global__ void node_fin_k(const float* __restrict__ gat, const float* __restrict__ gbias,
                           float* __restrict__ y, int total)
{
    int i = blockIdx.x * blockDim.x + threadIdx.x;
    if (i < total) y[i] = leaky(gat[i] + gbias[i & (HDIM - 1)]);
}

// ---------------------------------------------------------------------------
// acc[128] += sum_i y[i] * W1[i][c]   — streams the 327 MB W1 once (HBM-bound,
// ~14 us at 23.3 TB/s).  Thread t owns 4 columns: float4 (b128) coalesced
// loads, 8 rows in flight per block iteration, global_prefetch one grid-stride
// ahead, LDS fold, 128 atomics per block.
// ---------------------------------------------------------------------------
__global__ __launch_bounds__(256) void mv_w1_k(const float* __restrict__ y,
                                               const float* __restrict__ W1,
                                               float* __restrict__ acc, int rows)
{
    __shared__ float4 sdata[256];
    const int rg = threadIdx.x >> 5;          // row group 0..7
    const int c4 = (threadIdx.x & 31) * 4;    // first of 4 owned columns
    const int stride = gridDim.x * 8;
    float4 p = make_float4(0.f, 0.f, 0.f, 0.f);
    for (int i = blockIdx.x * 8 + rg; i < rows; i += stride) {
        // prefetch next grid-stride iteration of this stream (global_prefetch_b8)
        __builtin_prefetch(W1 + ((size_t)i + (size_t)stride) * HDIM + c4, 0, 0);
        float  yv = y[i];
        float4 w  = *(const float4*)(W1 + (size_t)i * HDIM + c4);
        p.x += yv * w.x; p.y += yv * w.y; p.z += yv * w.z; p.w += yv * w.w;
    }
    sdata[threadIdx.x] = p;
    __syncthreads();
    if (threadIdx.x < 32) {
        float4 t = sdata[threadIdx.x];
#pragma unroll
        for (int g = 1; g < 8; ++g) {
            float4 q = sdata[g * 32 + threadIdx.x];
            t.x += q.x; t.y += q.y; t.z += q.z; t.w += q.w;
        }
        atomicAdd(&acc[c4 + 0], t.x);
        atomicAdd(&acc[c4 + 1], t.y);
        atomicAdd(&acc[c4 + 2], t.z);
        atomicAdd(&acc[c4 + 3], t.w);
    }
}

__global__ void final_k(const float* __restrict__ acc, const float* __restrict__ b1,
                        const float* __restrict__ W2, const float* __restrict__ b2,
                        float* __restrict__ out)
{
    __shared__ float red[HDIM];
    int c = threadIdx.x;
    float v = acc[c] + b1[c];
    v = v > 0.f ? v : 0.f;               // relu
    red[c] = v * W2[c];
    __syncthreads();
    for (int s = HDIM / 2; s > 0; s >>= 1) {
        if (c < s) red[c] += red[c + s];
        __syncthreads();
    }
    if (c == 0) out[0] = red[0] + b2[0];
}

__global__ void fill_f32_k(float* p, float v, int n)
{
    int i = blockIdx.x * blockDim.x + threadIdx.x;
    if (i < n) p[i] = v;
}
__global__ void fill_u32_k(unsigned* p, unsigned v, int n)
{
    int i = blockIdx.x * blockDim.x + threadIdx.x;
    if (i < n) p[i] = v;
}

extern "C" void kernel_launch(void* const* d_in, const int* in_sizes, int n_in,
                              void* d_out, int out_size, void* d_ws, size_t ws_size,
                              hipStream_t stream)
{
    const float*     x        = (const float*)d_in[0];
    const long long* ei       = (const long long*)d_in[1];   // int64 [2, E]
    const float*     W_in     = (const float*)d_in[2];
    const float*     b_in     = (const float*)d_in[3];
    const float*     W_l      = (const float*)d_in[4];
    const float*     b_l      = (const float*)d_in[5];
    const float*     W_r      = (const float*)d_in[6];
    const float*     b_r      = (const float*)d_in[7];
    const float*     att      = (const float*)d_in[8];
    const float*     gat_bias = (const float*)d_in[9];
    const float*     W1       = (const float*)d_in[10];
    const float*     b1       = (const float*)d_in[11];
    const float*     W2       = (const float*)d_in[12];
    const float*     b2       = (const float*)d_in[13];
    float* out = (float*)d_out;

    const int IN_DIM = 1280;
    const int N  = in_sizes[0] / IN_DIM;   // 5000
    const int E  = in_sizes[1] / 2;        // 160000
    const int ET = E + N;
    const int NH = N * HDIM;

    float* ws = (float*)d_ws;
    size_t off = 0;
    float* h     = ws + off; off += (size_t)NH;
    float* xl    = ws + off; off += (size_t)NH;
    float* xr    = ws + off; off += (size_t)NH;
    float* gat   = ws + off; off += (size_t)NH;
    float* y     = ws + off; off += (size_t)NH;
    float* score = ws + off; off += (size_t)ET;
    float* exb   = ws + off; off += (size_t)ET;
    float* sum   = ws + off; off += (size_t)N;
    float* acc   = ws + off; off += (size_t)HDIM;
    unsigned* maxb = (unsigned*)(ws + off);

    // --- init scratch accumulators ---
    fill_f32_k<<<(NH + 255) / 256, 256, 0, stream>>>(gat, 0.f, NH);
    fill_f32_k<<<(N + 255) / 256, 256, 0, stream>>>(sum, 0.f, N);
    fill_f32_k<<<1, HDIM, 0, stream>>>(acc, 0.f, HDIM);
    fill_u32_k<<<(N + 255) / 256, 256, 0, stream>>>(maxb, ENC_NEG_INF, N);

    // --- dense GEMMs via fp32 WMMA (32-row strip per wave) ---
    int rowTiles   = (N + 15) / 16;        // 16-row tiles
    int gemmWaves  = (rowTiles + 1) / 2;   // 2 tiles per wave
    int gemmBlocks = (gemmWaves + 7) / 8;  // 8 waves per block
    gemm_wmma_f32<<<gemmBlocks, 256, 0, stream>>>(x, W_in, b_in, h, N, IN_DIM, 1); // relu
    gemm_wmma_f32<<<gemmBlocks, 256, 0, stream>>>(h, W_l, b_l, xl, N, HDIM, 0);
    gemm_wmma_f32<<<gemmBlocks, 256, 0, stream>>>(h, W_r, b_r, xr, N, HDIM, 0);

    // --- GATv2 edge phase (one wave per edge; gathers are L2-resident) ---
    int eWaveBlocks = (ET + 7) / 8;   // 8 waves per 256-thread block
    edge_score_k<<<eWaveBlocks, 256, 0, stream>>>(xl, xr, att, ei, E, N, score, maxb);
    edge_exp_k<<<(ET + 255) / 256, 256, 0, stream>>>(score, ei, E, N, maxb, exb, sum);
    edge_aggr_k<<<eWaveBlocks, 256, 0, stream>>>(xl, exb, sum, ei, E, N, gat);
    node_fin_k<<<(NH + 255) / 256, 256, 0, stream>>>(gat, gat_bias, y, NH);

    // --- hflat @ W1 (327 MB stream, HBM-bound) then tiny MLP head ---
    mv_w1_k<<<2048, 256, 0, stream>>>(y, W1, acc, NH);
    final_k<<<1, HDIM, 0, stream>>>(acc, b1, W2, b2, out);
}